// NAMTuring_55138790146731
// MI455X (gfx1250) — compile-verified
//
#include <hip/hip_runtime.h>

// ---------------------------------------------------------------------------
// NAM-Turing scan for MI455X (gfx1250, wave32).
//
// Math transformation: within a K=32-step block starting from tape0,
//   oldval_i = p_i^T tape0 + sum_{j<i} (p_i . p_j) * delta_j,  delta_j = w_j u_j
//   u_i      = v_i - oldval_i
// =>  (I + L) U = V - P_blk @ tape0,   L[i][j] = (p_i.p_j) * w_j  (j<i)
//     tape_new = tape0 + P_blk^T @ (W*U)
// Two dense f32 matmuls (V_WMMA_F32_16X16X4_F32) + a 32-deep forward
// substitution with channel-shared scalar coefficients. Exact f32 reassoc.
//
// Workspace (d_ws) layout, ~51 MB total:
//   aux    : [S*B] float4 (d0,d1,d2,w)          =  0.5 MB
//   Pg     : [B][S][TS] f32 pos history         = 16   MB
//   tape_g : [TS][B][C] f32 final tape          = 32   MB
// ---------------------------------------------------------------------------

#define S_DIM   512
#define B_DIM   64
#define C_DIM   1024
#define V_DIM   10
#define TS      128
#define KB      32
#define NBLK    (S_DIM / KB)        // 16
#define C_TILE  256
#define PSTR    132                 // P block LDS row stride (pad: conflict-free col reads)
#define RSTR    260                 // RHS/Delta LDS row stride
#define TSTR    260                 // tape LDS row stride
#define SMEM_C  ((TS*TSTR + KB*PSTR + KB*KB + KB*RSTR + KB) * 4)  // 187520 B

typedef float v2f __attribute__((ext_vector_type(2)));
typedef float v8f __attribute__((ext_vector_type(8)));

// 16x16x4 f32 WMMA: A 16x4 (2 VGPR), B 4x16 (2 VGPR), C/D 16x16 (8 VGPR).
__device__ __forceinline__ v8f wmma4(v2f a, v2f b, v8f c) {
  return __builtin_amdgcn_wmma_f32_16x16x4_f32(
      /*neg_a=*/false, a, /*neg_b=*/false, b,
      /*c_mod=*/(short)0, c, /*reuse_a=*/false, /*reuse_b=*/false);
}

__device__ __forceinline__ float wred32(float x) {
#pragma unroll
  for (int o = 16; o > 0; o >>= 1) x += __shfl_xor(x, o, 32);
  return x;
}

// ---------------------------------------------------------------------------
// Kernel A: gating.  aux[r] = (softmax3(v.Wd+bd), sigmoid(v.Wr[1]+br[1]))
// One wave per row r = s*B+b; memory-bound (reads values once, 128 MB).
// ---------------------------------------------------------------------------
__global__ void __launch_bounds__(256) nam_gate_kernel(
    const float* __restrict__ values, const float* __restrict__ Wd,
    const float* __restrict__ bd, const float* __restrict__ Wr,
    const float* __restrict__ br, float4* __restrict__ aux) {
  const int lane = threadIdx.x & 31;
  const int wave = threadIdx.x >> 5;
  const int row  = blockIdx.x * 8 + wave;           // < S*B
  const float* vrow = values + (size_t)row * C_DIM;
  float a0 = 0.f, a1 = 0.f, a2 = 0.f, a3 = 0.f;
#pragma unroll 4
  for (int k = 0; k < C_DIM / 32; ++k) {
    const int c = lane + 32 * k;
    const float x = vrow[c];
    a0 = fmaf(x, Wd[0 * C_DIM + c], a0);
    a1 = fmaf(x, Wd[1 * C_DIM + c], a1);
    a2 = fmaf(x, Wd[2 * C_DIM + c], a2);
    a3 = fmaf(x, Wr[1 * C_DIM + c], a3);
  }
  a0 = wred32(a0); a1 = wred32(a1); a2 = wred32(a2); a3 = wred32(a3);
  if (lane == 0) {
    const float z0 = a0 + bd[0], z1 = a1 + bd[1], z2 = a2 + bd[2];
    const float m  = fmaxf(z0, fmaxf(z1, z2));
    const float e0 = __expf(z0 - m), e1 = __expf(z1 - m), e2 = __expf(z2 - m);
    const float inv = 1.f / (e0 + e1 + e2);
    const float w  = 1.f / (1.f + __expf(-(a3 + br[1])));
    aux[row] = make_float4(e0 * inv, e1 * inv, e2 * inv, w);
  }
}

// ---------------------------------------------------------------------------
// Kernel B: pos evolution (independent of tape). One 128-thread block / batch.
// Pg[b][s][l] = pos before step s.  pos0 = one-hot(0).
// ---------------------------------------------------------------------------
__global__ void __launch_bounds__(TS) nam_pos_kernel(
    const float4* __restrict__ aux, float* __restrict__ Pg) {
  const int b = blockIdx.x;
  const int l = threadIdx.x;
  __shared__ float cur[TS];
  float pos = (l == 0) ? 1.f : 0.f;
  float* Pb = Pg + (size_t)b * S_DIM * TS;
  for (int s = 0; s < S_DIM; ++s) {
    Pb[s * TS + l] = pos;
    cur[l] = pos;
    __syncthreads();
    const float4 d = aux[s * B_DIM + b];
    // roll(pos,-1)[l]=pos[l+1] ; roll(pos,+1)[l]=pos[l-1]
    const float up = cur[(l + 1) & (TS - 1)];
    const float dn = cur[(l - 1) & (TS - 1)];
    pos = fmaf(up, d.x, fmaf(pos, d.y, dn * d.z));
    __syncthreads();
  }
}

// ---------------------------------------------------------------------------
// Kernel C: blocked scan. One WG per (batch, 256-channel tile); 8 waves.
// tape tile [128][256] f32 resident in LDS for all 512 steps.
// ---------------------------------------------------------------------------
__global__ void __launch_bounds__(256, 1) nam_scan_kernel(
    const float* __restrict__ values, const float4* __restrict__ aux,
    const float* __restrict__ Pg, float* __restrict__ tape_g) {
  const int b    = blockIdx.x >> 2;
  const int ct   = blockIdx.x & 3;
  const int c0   = ct * C_TILE;
  const int t    = threadIdx.x;
  const int lane = t & 31;
  const int wave = t >> 5;

  extern __shared__ float smem[];
  float* tape = smem;                        // [TS][TSTR]
  float* Pp   = tape + TS * TSTR;            // [KB][PSTR]
  float* Lw   = Pp + KB * PSTR;              // [KB][KB]
  float* Rm   = Lw + KB * KB;                // [KB][RSTR] (RHS, then Delta)
  float* wb   = Rm + KB * RSTR;              // [KB]

  for (int i = t; i < TS * TSTR; i += 256) tape[i] = 0.f;
  __syncthreads();

  const float* Pb = Pg + (size_t)b * S_DIM * TS;

  for (int blk = 0; blk < NBLK; ++blk) {
    const int base = blk * KB;

    // ---- stage P block + write probs into LDS -------------------------
    for (int i = t; i < KB * TS; i += 256) {
      const int r = i >> 7, cc = i & (TS - 1);
      Pp[r * PSTR + cc] = Pb[(base + r) * TS + cc];
    }
    if (t < KB) wb[t] = aux[(base + t) * B_DIM + b].w;
    if (blk + 1 < NBLK) {  // prefetch next block's P and values slab into L2
      __builtin_prefetch((const char*)(Pb + (size_t)(base + KB) * TS) + t * 64, 0, 1);
      __builtin_prefetch((const char*)(values +
          ((size_t)(base + KB) * B_DIM + b) * C_DIM + c0) + t * 64, 0, 1);
    }
    __syncthreads();

    // ---- Lw[i][j] = (p_i . p_j) * w_j ---------------------------------
    for (int p = t; p < KB * KB; p += 256) {
      const int i = p >> 5, j = p & 31;
      float g = 0.f;
#pragma unroll 4
      for (int l2 = 0; l2 < TS; ++l2) g = fmaf(Pp[i * PSTR + l2], Pp[j * PSTR + l2], g);
      Lw[p] = g * wb[j];
    }
    __syncthreads();

    // ---- matmul1: Rm = P_blk[32,128] x tape[128,C_TILE]  (WMMA) -------
    {
      const int hi = lane >> 4, ml = lane & 15, k0 = hi * 2, ro = hi * 8;
      const int nj0 = wave * 2;
      v8f acc[2][2];
#pragma unroll
      for (int mi = 0; mi < 2; ++mi)
#pragma unroll
        for (int q = 0; q < 2; ++q) acc[mi][q] = (v8f){0,0,0,0,0,0,0,0};
      for (int kk = 0; kk < TS / 4; ++kk) {
        const int kb = 4 * kk + k0;
        v2f af[2], bf[2];
#pragma unroll
        for (int mi = 0; mi < 2; ++mi) {
          af[mi].x = Pp[(16 * mi + ml) * PSTR + kb];
          af[mi].y = Pp[(16 * mi + ml) * PSTR + kb + 1];
        }
#pragma unroll
        for (int q = 0; q < 2; ++q) {
          bf[q].x = tape[kb * TSTR + 16 * (nj0 + q) + ml];
          bf[q].y = tape[(kb + 1) * TSTR + 16 * (nj0 + q) + ml];
        }
#pragma unroll
        for (int mi = 0; mi < 2; ++mi)
#pragma unroll
          for (int q = 0; q < 2; ++q) acc[mi][q] = wmma4(af[mi], bf[q], acc[mi][q]);
      }
#pragma unroll
      for (int mi = 0; mi < 2; ++mi)
#pragma unroll
        for (int q = 0; q < 2; ++q)
#pragma unroll
          for (int r = 0; r < 8; ++r)
            Rm[(16 * mi + r + ro) * RSTR + 16 * (nj0 + q) + ml] = acc[mi][q][r];
    }
    __syncthreads();

    // ---- forward substitution (thread t owns channel c0+t) ------------
    {
      const float* vcol = values + ((size_t)base * B_DIM + b) * C_DIM + c0 + t;
      float u[KB];
#pragma unroll
      for (int i = 0; i < KB; ++i) {
        float r = vcol[(size_t)i * B_DIM * C_DIM] - Rm[i * RSTR + t];
#pragma unroll
        for (int j = 0; j < i; ++j) r = fmaf(-Lw[i * KB + j], u[j], r);
        u[i] = r;
        Rm[i * RSTR + t] = wb[i] * r;   // delta_i
      }
    }
    __syncthreads();

    // ---- matmul2: tape += P_blk^T[128,32] x Delta[32,C_TILE]  (WMMA) --
    {
      const int hi = lane >> 4, ml = lane & 15, k0 = hi * 2, ro = hi * 8;
      const int nj0 = wave * 2;
      for (int mt = 0; mt < 8; ++mt) {
#pragma unroll
        for (int q = 0; q < 2; ++q) {
          const int nc = 16 * (nj0 + q);
          v8f cfr;
#pragma unroll
          for (int r = 0; r < 8; ++r) cfr[r] = tape[(16 * mt + r + ro) * TSTR + nc + ml];
#pragma unroll
          for (int kk = 0; kk < KB / 4; ++kk) {
            const int kb = 4 * kk + k0;
            v2f af, bf;
            af.x = Pp[kb * PSTR + 16 * mt + ml];        // A[l][i] = P[i][l]
            af.y = Pp[(kb + 1) * PSTR + 16 * mt + ml];
            bf.x = Rm[kb * RSTR + nc + ml];
            bf.y = Rm[(kb + 1) * RSTR + nc + ml];
            cfr = wmma4(af, bf, cfr);
          }
#pragma unroll
          for (int r = 0; r < 8; ++r) tape[(16 * mt + r + ro) * TSTR + nc + ml] = cfr[r];
        }
      }
    }
    __syncthreads();
  }

  // ---- spill final tape tile to global [TS][B][C] ---------------------
  for (int l = 0; l < TS; ++l)
    tape_g[((size_t)l * B_DIM + b) * C_DIM + c0 + t] = tape[l * TSTR + t];
}

// ---------------------------------------------------------------------------
// Kernel D: LayerNorm + [C]x[C,10] decode. One wave per (t,b) row.
// ---------------------------------------------------------------------------
__global__ void __launch_bounds__(256) nam_decode_kernel(
    const float* __restrict__ tape_g, const float* __restrict__ ln_g,
    const float* __restrict__ ln_b, const float* __restrict__ Wo,
    const float* __restrict__ bo, float* __restrict__ out) {
  const int lane = threadIdx.x & 31;
  const int wave = threadIdx.x >> 5;
  const int idx  = blockIdx.x * 8 + wave;            // < TS*B
  const float* row = tape_g + (size_t)idx * C_DIM;
  float x[C_DIM / 32];
  float s = 0.f, s2 = 0.f;
#pragma unroll
  for (int k = 0; k < C_DIM / 32; ++k) {
    const float xv = row[lane + 32 * k];
    x[k] = xv;
    s += xv;
    s2 = fmaf(xv, xv, s2);
  }
  s  = wred32(s);
  s2 = wred32(s2);
  const float mu  = s * (1.f / C_DIM);
  const float var = s2 * (1.f / C_DIM) - mu * mu;
  const float rs  = rsqrtf(var + 1e-5f);
  float acc[V_DIM];
#pragma unroll
  for (int v = 0; v < V_DIM; ++v) acc[v] = 0.f;
#pragma unroll 2
  for (int k = 0; k < C_DIM / 32; ++k) {
    const int c = lane + 32 * k;
    const float h = fmaf((x[k] - mu) * rs, ln_g[c], ln_b[c]);
#pragma unroll
    for (int v = 0; v < V_DIM; ++v) acc[v] = fmaf(h, Wo[v * C_DIM + c], acc[v]);
  }
#pragma unroll
  for (int v = 0; v < V_DIM; ++v) acc[v] = wred32(acc[v]);
  if (lane == 0) {
#pragma unroll
    for (int v = 0; v < V_DIM; ++v) out[(size_t)idx * V_DIM + v] = acc[v] + bo[v];
  }
}

// ---------------------------------------------------------------------------
extern "C" void kernel_launch(void* const* d_in, const int* in_sizes, int n_in,
                              void* d_out, int out_size, void* d_ws, size_t ws_size,
                              hipStream_t stream) {
  (void)in_sizes; (void)n_in; (void)out_size; (void)ws_size;
  const float* values = (const float*)d_in[0];
  const float* Wd     = (const float*)d_in[1];
  const float* bd     = (const float*)d_in[2];
  const float* Wr     = (const float*)d_in[3];
  const float* br     = (const float*)d_in[4];
  const float* ln_g   = (const float*)d_in[5];
  const float* ln_b   = (const float*)d_in[6];
  const float* Wo     = (const float*)d_in[7];
  const float* bo     = (const float*)d_in[8];
  float* out = (float*)d_out;

  char* ws = (char*)d_ws;
  float4* aux   = (float4*)ws;                                       // S*B float4
  float*  Pg    = (float*)(ws + (size_t)S_DIM * B_DIM * 16);         // B*S*TS f32
  float*  tapeg = (float*)(ws + (size_t)S_DIM * B_DIM * 16 +
                                (size_t)B_DIM * S_DIM * TS * 4);     // TS*B*C f32

  (void)hipFuncSetAttribute((const void*)nam_scan_kernel,
                            hipFuncAttributeMaxDynamicSharedMemorySize, SMEM_C);

  nam_gate_kernel<<<(S_DIM * B_DIM) / 8, 256, 0, stream>>>(values, Wd, bd, Wr, br, aux);
  nam_pos_kernel<<<B_DIM, TS, 0, stream>>>(aux, Pg);
  nam_scan_kernel<<<B_DIM * (C_DIM / C_TILE), 256, SMEM_C, stream>>>(values, aux, Pg, tapeg);
  nam_decode_kernel<<<(TS * B_DIM) / 8, 256, 0, stream>>>(tapeg, ln_g, ln_b, Wo, bo, out);
}